// GCN_large_6201932775762
// MI455X (gfx1250) — compile-verified
//
#include <hip/hip_runtime.h>
#include <hip/hip_bf16.h>

// ---------------------------------------------------------------------------
// GCN (5x GCNConv + linear), C = 128, fp32.
// Strategy (MI455X):
//  * All dense GEMMs via V_WMMA_F32_16X16X4_F32 (exact fp32, K-loop of 32).
//  * Edge scatter: 1 wave32 per edge, 32 lanes x float4 = 128 channels,
//    relaxed agent-scope fp32 atomic adds (resolve at L2; the 25.6MB node
//    matrix fits in the 192MB L2, so gather+atomics stay on-chip).
//  * Self-loop term folded into output init: out = b + dinv[i]^2 * (hW)[i].
//  * ReLU fused into next GEMM's A-load (relu_in) / final epilogue (relu_out).
//  * Buffers phased so no kernel ever reads and writes the same array
//    (keeps __restrict__ valid everywhere).
// ---------------------------------------------------------------------------

typedef __attribute__((ext_vector_type(2))) float v2f;
typedef __attribute__((ext_vector_type(8))) float v8f;

#define C128 128

// ---------------- degree / norm ----------------

__global__ void k_init_deg(float* deg, int n) {
    int i = blockIdx.x * blockDim.x + threadIdx.x;
    if (i < n) deg[i] = 1.0f;  // self-loop contributes 1 to every dst degree
}

__global__ void k_deg_accum(const int* __restrict__ dst, float* deg, int ne) {
    int e = blockIdx.x * blockDim.x + threadIdx.x;
    if (e < ne) {
        (void)__hip_atomic_fetch_add(&deg[dst[e]], 1.0f,
                                     __ATOMIC_RELAXED, __HIP_MEMORY_SCOPE_AGENT);
    }
}

__global__ void k_rsqrt(float* deg_dinv, int n) {
    int i = blockIdx.x * blockDim.x + threadIdx.x;
    if (i < n) {
        float d = deg_dinv[i];
        deg_dinv[i] = (d > 0.0f) ? rsqrtf(d) : 0.0f;
    }
}

// ---------------- GEMM: Out = [relu_in](A) @ W  (+bias, +relu_out) ----------
// A: [nrows, 128], W: [128, 128] row-major, Out: [nrows, 128]. A != Out.
// Block: 256 threads = 8 waves; each wave does 16 rows x 128 cols.

__global__ __launch_bounds__(256) void k_gemm128(
    const float* __restrict__ A, const float* __restrict__ W,
    const float* __restrict__ bias, float* __restrict__ Out,
    int nrows, int relu_in, int add_bias, int relu_out)
{
    // W staged in LDS, pair-swizzled so (W[k][n], W[k+1][n]) for even k are
    // adjacent: sWp[(k>>1)*256 + n*2 + (k&1)]  -> one 8B read per B fragment.
    __shared__ float sWp[C128 * C128];
    for (int i = threadIdx.x; i < C128 * C128; i += 256) {
        int k = i >> 7, n = i & 127;
        sWp[(k >> 1) * 256 + n * 2 + (k & 1)] = W[i];
    }
    __syncthreads();

    const int wave = threadIdx.x >> 5;
    const int lane = threadIdx.x & 31;
    const int half = lane >> 4;       // 0: K={0,1}, 1: K={2,3} within 4-step
    const int lrow16 = lane & 15;
    const int row0 = blockIdx.x * 128 + wave * 16;

    // Preload A fragments for all K (clamped row for OOB lanes; those output
    // rows are never stored, and garbage rows cannot pollute valid rows since
    // A-row M only feeds D-row M). No early return: WMMA needs full EXEC.
    int lrow = row0 + lrow16;
    int lrowc = (lrow < nrows) ? lrow : (nrows - 1);
    const float* Arow = A + (size_t)lrowc * C128;
    v2f afrag[32];
#pragma unroll
    for (int kk = 0; kk < 32; kk++) {
        float a0 = Arow[kk * 4 + half * 2 + 0];
        float a1 = Arow[kk * 4 + half * 2 + 1];
        if (relu_in) { a0 = fmaxf(a0, 0.0f); a1 = fmaxf(a1, 0.0f); }
        afrag[kk] = (v2f){a0, a1};
    }

#pragma unroll
    for (int nt = 0; nt < 8; nt++) {
        const int col = nt * 16 + lrow16;
        v8f acc = {0.f, 0.f, 0.f, 0.f, 0.f, 0.f, 0.f, 0.f};
#pragma unroll
        for (int kk = 0; kk < 32; kk++) {
            const float* bp = &sWp[(size_t)(2 * kk + half) * 256 + col * 2];
            v2f b = (v2f){bp[0], bp[1]};
            acc = __builtin_amdgcn_wmma_f32_16x16x4_f32(
                false, afrag[kk], false, b, (short)0, acc, false, false);
        }
        // C/D layout: VGPR j, lanes 0-15 -> M=j, lanes 16-31 -> M=j+8; N=lane%16
#pragma unroll
        for (int j = 0; j < 8; j++) {
            int r = row0 + half * 8 + j;
            if (r < nrows) {
                float v = acc[j];
                if (add_bias) v += bias[col];
                if (relu_out) v = fmaxf(v, 0.0f);
                Out[(size_t)r * C128 + col] = v;
            }
        }
    }
}

// ---------------- self-loop + bias init: out = b + dinv[i]^2 * t ------------

__global__ void k_selfloop_init(const float* __restrict__ tmp,
                                const float* __restrict__ dinv,
                                const float* __restrict__ bias,
                                float* __restrict__ out, int n)
{
    int i = blockIdx.x * blockDim.x + threadIdx.x;   // over n*32 float4s
    if (i < n * 32) {
        int r = i >> 5;
        int c4 = i & 31;
        float di = dinv[r];
        float s = di * di;
        float4 t = ((const float4*)tmp)[i];
        float4 bv = ((const float4*)bias)[c4];
        float4 o;
        o.x = bv.x + s * t.x; o.y = bv.y + s * t.y;
        o.z = bv.z + s * t.z; o.w = bv.w + s * t.w;
        ((float4*)out)[i] = o;
    }
}

// ---------------- edge scatter: out[dst] += dinv[s]*dinv[d]*tmp[src] --------
// One wave per edge; lane l covers channels [4l, 4l+3].

__global__ __launch_bounds__(256) void k_scatter(
    const int* __restrict__ src, const int* __restrict__ dst,
    const float* __restrict__ dinv, const float* __restrict__ tmp,
    float* __restrict__ out, int ne)
{
    int wid = (blockIdx.x * blockDim.x + threadIdx.x) >> 5;
    int lane = threadIdx.x & 31;
    if (wid >= ne) return;
    int s = src[wid];
    int d = dst[wid];
    float w = dinv[s] * dinv[d];
    float4 v = ((const float4*)(tmp + (size_t)s * C128))[lane];
    float* po = out + (size_t)d * C128 + lane * 4;
    (void)__hip_atomic_fetch_add(po + 0, v.x * w, __ATOMIC_RELAXED, __HIP_MEMORY_SCOPE_AGENT);
    (void)__hip_atomic_fetch_add(po + 1, v.y * w, __ATOMIC_RELAXED, __HIP_MEMORY_SCOPE_AGENT);
    (void)__hip_atomic_fetch_add(po + 2, v.z * w, __ATOMIC_RELAXED, __HIP_MEMORY_SCOPE_AGENT);
    (void)__hip_atomic_fetch_add(po + 3, v.w * w, __ATOMIC_RELAXED, __HIP_MEMORY_SCOPE_AGENT);
}

// ---------------------------------------------------------------------------

extern "C" void kernel_launch(void* const* d_in, const int* in_sizes, int n_in,
                              void* d_out, int out_size, void* d_ws, size_t ws_size,
                              hipStream_t stream)
{
    const float* x  = (const float*)d_in[0];
    const int*   ei = (const int*)d_in[1];        // [2, E] int32
    // d_in[2] = edge_attr (unused by reference path)
    const float* W[6] = {(const float*)d_in[3], (const float*)d_in[5],
                         (const float*)d_in[7], (const float*)d_in[9],
                         (const float*)d_in[11], (const float*)d_in[13]};
    const float* B[6] = {(const float*)d_in[4], (const float*)d_in[6],
                         (const float*)d_in[8], (const float*)d_in[10],
                         (const float*)d_in[12], (const float*)d_in[14]};

    const int N = in_sizes[0] / C128;
    const int E = in_sizes[1] / 2;
    const int* src = ei;
    const int* dst = ei + E;

    // Workspace: dinv[N] | bufT[N*128] | bufB[N*128]  (~52 MB); d_out doubles
    // as one of the ping-pong h buffers.
    float* dinv = (float*)d_ws;
    size_t nAlign = ((size_t)N + 63) & ~(size_t)63;
    float* bufT = dinv + nAlign;
    float* bufB = bufT + (size_t)N * C128;
    float* outF = (float*)d_out;

    dim3 blk(256);
    dim3 gN((N + 255) / 256);
    dim3 gE((E + 255) / 256);
    dim3 gGemm((N + 127) / 128);
    dim3 gInit((N * 32 + 255) / 256);
    dim3 gScat((E + 7) / 8);          // 8 waves (edges) per block

    // Degrees -> dinv (in place)
    k_init_deg<<<gN, blk, 0, stream>>>(dinv, N);
    k_deg_accum<<<gE, blk, 0, stream>>>(dst, dinv, E);
    k_rsqrt<<<gN, blk, 0, stream>>>(dinv, N);

    // 5 GCN layers; h ping-pongs bufB <-> outF, phased so the final linear
    // reads bufB and writes d_out (no kernel aliases input with output).
    const float* hin = x;
    float* houts[5] = {bufB, outF, bufB, outF, bufB};
    for (int l = 0; l < 5; l++) {
        k_gemm128<<<gGemm, blk, 0, stream>>>(hin, W[l], nullptr, bufT, N,
                                             /*relu_in=*/(l > 0) ? 1 : 0,
                                             /*add_bias=*/0, /*relu_out=*/0);
        k_selfloop_init<<<gInit, blk, 0, stream>>>(bufT, dinv, B[l], houts[l], N);
        k_scatter<<<gScat, blk, 0, stream>>>(src, dst, dinv, bufT, houts[l], E);
        hin = houts[l];
    }

    // Final linear: out = relu(relu(h5) @ Wl + bl);  h5 = bufB, out = d_out.
    k_gemm128<<<gGemm, blk, 0, stream>>>(hin, W[5], B[5], outF, N,
                                         /*relu_in=*/1, /*add_bias=*/1,
                                         /*relu_out=*/1);
}